// ElectricOverflow_69879117906279
// MI455X (gfx1250) — compile-verified
//
#include <hip/hip_runtime.h>

// Problem constants (from reference)
#define NN     2000000      // NUM_MOVABLE + NUM_FILLER
#define NBX_C  512
#define NBY_C  512
#define NBINS  (NBX_C * NBY_C)   // 262144

typedef __attribute__((ext_vector_type(2))) float v2f;
typedef __attribute__((ext_vector_type(8))) float v8f;

// Guaranteed hardware FP32 atomic add (no CAS-loop fallback).
__device__ __forceinline__ void gatomic_add_f32(float* addr, float val) {
    asm volatile("global_atomic_add_f32 %0, %1, off"
                 :: "v"(addr), "v"(val)
                 : "memory");
}

// ---------------------------------------------------------------------------
// Kernel 1: initialize density map (copy initial map) and zero outputs.
// Must run every call: graph replay leaves last call's map in d_ws.
// ---------------------------------------------------------------------------
__global__ void eo_init_kernel(const float* __restrict__ init_map,
                               float* __restrict__ dm,
                               float* __restrict__ out) {
    int i = blockIdx.x * blockDim.x + threadIdx.x;
    if (i < NBINS) dm[i] = init_map[i];
    if (i == 0) { out[0] = 0.0f; out[1] = 0.0f; }
}

// ---------------------------------------------------------------------------
// Kernel 2: scatter 3x3 overlap contributions with HW f32 atomics.
// Map (1 MB) is L2-resident on MI455X (192 MB L2) -> atomics stay in L2.
// ---------------------------------------------------------------------------
__global__ void eo_scatter_kernel(const float* __restrict__ pos,
                                  const float* __restrict__ nsx,
                                  const float* __restrict__ nsy,
                                  float* __restrict__ dm) {
    int i = blockIdx.x * blockDim.x + threadIdx.x;
    if (i >= NN) return;

    const float szx = nsx[i];
    const float szy = nsy[i];
    const float cx  = pos[i]      + 0.5f * szx;
    const float cy  = pos[NN + i] + 0.5f * szy;
    // TARGET_HALF_X = TARGET_HALF_Y = 1.0 (stretch 2.0 * bin 1.0 * 0.5)
    const float sx = fmaxf(1.0f, 0.5f * szx);
    const float sy = fmaxf(1.0f, 0.5f * szy);
    const float scale = 0.25f * szx * szy / (sx * sy);

    const int lbx = (int)floorf(cx - sx);   // XL=0, BIN_X=1
    const int lby = (int)floorf(cy - sy);   // YL=0, BIN_Y=1

    float ovx[3], ovy[3];
    int   bxc[3], byc[3];
#pragma unroll
    for (int k = 0; k < 3; ++k) {
        int bx = lbx + k;
        float ox = fminf(cx + sx, (float)(bx + 1)) - fmaxf(cx - sx, (float)bx);
        ox = fmaxf(ox, 0.0f);
        if (bx < 0 || bx >= NBX_C) ox = 0.0f;
        ovx[k] = ox;
        bxc[k] = min(max(bx, 0), NBX_C - 1);

        int by = lby + k;
        float oy = fminf(cy + sy, (float)(by + 1)) - fmaxf(cy - sy, (float)by);
        oy = fmaxf(oy, 0.0f);
        if (by < 0 || by >= NBY_C) oy = 0.0f;
        ovy[k] = oy;
        byc[k] = min(max(by, 0), NBY_C - 1);
    }

#pragma unroll
    for (int kx = 0; kx < 3; ++kx) {
        const float sox = scale * ovx[kx];
        const int   row = bxc[kx] * NBY_C;
#pragma unroll
        for (int ky = 0; ky < 3; ++ky) {
            const float c = sox * ovy[ky];
            if (c != 0.0f) gatomic_add_f32(dm + row + byc[ky], c);
        }
    }
}

// ---------------------------------------------------------------------------
// Kernel 3: reduction. 65536 threads x 4 coalesced cells.
// Wave-level SUM via V_WMMA_F32_16X16X4_F32 with B = ones:
//   A[m][0] = partial(lane m), A[m][2] = partial(lane m+16), rest 0
//   D[m][n] = partial[m] + partial[m+16]  (row-sums, replicated over n)
// Each lane sums its 8 D registers (covers 8 of 16 row-sums for its column);
// lanes L and L^16 hold complementary halves -> one shfl_xor finishes it.
// EXEC is all-ones here (no divergence), as WMMA requires.
// ---------------------------------------------------------------------------
__device__ __forceinline__ float wave_sum_wmma(float v) {
    v2f a; a.x = v;    a.y = 0.0f;
    v2f b; b.x = 1.0f; b.y = 1.0f;   // all-ones B: layout-independent
    v8f c = {};
    c = __builtin_amdgcn_wmma_f32_16x16x4_f32(
            /*neg_a=*/false, a, /*neg_b=*/false, b,
            /*c_mod=*/(short)0, c, /*reuse_a=*/false, /*reuse_b=*/false);
    float s = c[0] + c[1] + c[2] + c[3] + c[4] + c[5] + c[6] + c[7];
    s += __shfl_xor(s, 16);
    return s;
}

__device__ __forceinline__ float wave_max(float v) {
#pragma unroll
    for (int o = 16; o > 0; o >>= 1) v = fmaxf(v, __shfl_xor(v, o));
    return v;
}

__global__ void eo_reduce_kernel(const float* __restrict__ dm,
                                 float* __restrict__ out) {
    const int t = blockIdx.x * 256 + threadIdx.x;   // 65536 threads total
    float lsum = 0.0f, lmax = 0.0f;
#pragma unroll
    for (int k = 0; k < 4; ++k) {
        const float v = dm[t + k * 65536];
        lsum += fmaxf(v - 1.0f, 0.0f);   // TARGET_DENSITY * bin_area = 1
        lmax  = fmaxf(lmax, v);
    }

    const float ws = wave_sum_wmma(lsum);
    const float wm = wave_max(lmax);

    __shared__ float ssum[8];
    __shared__ float smax[8];
    const int wid  = threadIdx.x >> 5;
    const int lane = threadIdx.x & 31;
    if (lane == 0) { ssum[wid] = ws; smax[wid] = wm; }
    __syncthreads();

    if (threadIdx.x == 0) {
        float ts = 0.0f, tm = 0.0f;
#pragma unroll
        for (int w = 0; w < 8; ++w) {
            ts += ssum[w];
            tm  = fmaxf(tm, smax[w]);
        }
        gatomic_add_f32(&out[0], ts);
        // map values are >= 0 -> IEEE bits are order-isomorphic to uints
        atomicMax(reinterpret_cast<unsigned int*>(&out[1]), __float_as_uint(tm));
    }
}

// ---------------------------------------------------------------------------
extern "C" void kernel_launch(void* const* d_in, const int* in_sizes, int n_in,
                              void* d_out, int out_size, void* d_ws, size_t ws_size,
                              hipStream_t stream) {
    const float* pos      = (const float*)d_in[0];   // 2N
    const float* nsx      = (const float*)d_in[1];   // N
    const float* nsy      = (const float*)d_in[2];   // N
    const float* init_map = (const float*)d_in[3];   // NBX*NBY
    float*       out      = (float*)d_out;           // [density_cost, max_density]
    float*       dm       = (float*)d_ws;            // 512*512 f32 density map

    eo_init_kernel<<<(NBINS + 255) / 256, 256, 0, stream>>>(init_map, dm, out);
    eo_scatter_kernel<<<(NN + 255) / 256, 256, 0, stream>>>(pos, nsx, nsy, dm);
    eo_reduce_kernel<<<256, 256, 0, stream>>>(dm, out);
}